// SGConvNet_1layer_88553635709231
// MI455X (gfx1250) — compile-verified
//
#include <hip/hip_runtime.h>
#include <math.h>

#define N_NODES 100000
#define N_EDGES 3200000
#define D_IN    256
#define D_HID   32
#define N_CLS   40
#define NBLK    ((N_NODES + 255) / 256)   // 391 scan blocks

typedef float v2f __attribute__((ext_vector_type(2)));
typedef float v8f __attribute__((ext_vector_type(8)));

// ---------------------------------------------------------------- init
__global__ void k_init(float* __restrict__ deg, int* __restrict__ cnt) {
    int i = blockIdx.x * blockDim.x + threadIdx.x;
    if (i < N_NODES) { deg[i] = 1.0f; cnt[i] = 0; }   // self-loop weight 1
}

// per-edge: weighted in-degree (float) + edge count (int) by destination
__global__ void k_acc(const long long* __restrict__ col,
                      const float* __restrict__ w,
                      float* __restrict__ deg, int* __restrict__ cnt) {
    int e = blockIdx.x * blockDim.x + threadIdx.x;
    if (e < N_EDGES) {
        int d = (int)col[e];
        atomicAdd(&deg[d], w[e]);
        atomicAdd(&cnt[d], 1);
    }
}

__global__ void k_dis(const float* __restrict__ deg, float* __restrict__ dis) {
    int i = blockIdx.x * blockDim.x + threadIdx.x;
    if (i < N_NODES) {
        float d = deg[i];
        dis[i] = (d > 0.0f) ? (1.0f / sqrtf(d)) : 0.0f;
    }
}

// ---------------------------------------------------------------- scan (CSR)
__global__ void __launch_bounds__(256)
k_blocksum(const int* __restrict__ cnt, int* __restrict__ bsum) {
    __shared__ int s[256];
    int t = threadIdx.x;
    int i = blockIdx.x * 256 + t;
    s[t] = (i < N_NODES) ? cnt[i] : 0;
    __syncthreads();
    for (int off = 128; off > 0; off >>= 1) {
        if (t < off) s[t] += s[t + off];
        __syncthreads();
    }
    if (t == 0) bsum[blockIdx.x] = s[0];
}

// single block: in-place exclusive scan of the 391 block sums (padded to 512)
__global__ void __launch_bounds__(512)
k_scanblocks(int* __restrict__ bsum) {
    __shared__ int s[512];
    int t = threadIdx.x;
    int v = (t < NBLK) ? bsum[t] : 0;
    s[t] = v;
    __syncthreads();
    for (int off = 1; off < 512; off <<= 1) {
        int x = (t >= off) ? s[t - off] : 0;
        __syncthreads();
        s[t] += x;
        __syncthreads();
    }
    if (t < NBLK) bsum[t] = s[t] - v;   // exclusive
}

__global__ void __launch_bounds__(256)
k_rowptr(const int* __restrict__ cnt, const int* __restrict__ bsum,
         int* __restrict__ rowptr, int* __restrict__ pos) {
    __shared__ int s[256];
    int t = threadIdx.x;
    int i = blockIdx.x * 256 + t;
    int v = (i < N_NODES) ? cnt[i] : 0;
    s[t] = v;
    __syncthreads();
    for (int off = 1; off < 256; off <<= 1) {
        int x = (t >= off) ? s[t - off] : 0;
        __syncthreads();
        s[t] += x;
        __syncthreads();
    }
    int excl = bsum[blockIdx.x] + s[t] - v;
    if (i < N_NODES) { rowptr[i] = excl; pos[i] = excl; }
    if (i == 0) rowptr[N_NODES] = N_EDGES;
}

// fill CSR slots; fuse edge normalization dis[row]*w*dis[col]
__global__ void __launch_bounds__(256)
k_fill(const long long* __restrict__ row, const long long* __restrict__ col,
       const float* __restrict__ w, const float* __restrict__ dis,
       int* __restrict__ pos, int* __restrict__ esrc, float* __restrict__ enorm) {
    int e = blockIdx.x * blockDim.x + threadIdx.x;
    if (e < N_EDGES) {
        int r = (int)row[e];
        int d = (int)col[e];
        float nv = dis[r] * w[e] * dis[d];
        int idx = atomicAdd(&pos[d], 1);
        esrc[idx]  = r;
        enorm[idx] = nv;
    }
}

// ---------------------------------------------------------------- hop (gather)
// One block per destination node; thread t owns feature t. No atomics.
// out[dst,t] = dis[dst]^2 * in[dst,t] + sum_j enorm[j] * in[esrc[j], t]
__global__ void __launch_bounds__(256)
k_hop(const int* __restrict__ rowptr, const int* __restrict__ esrc,
      const float* __restrict__ enorm, const float* __restrict__ dis,
      const float* __restrict__ in, float* __restrict__ out) {
    const int dst = blockIdx.x;
    const int t   = threadIdx.x;
    const int beg = rowptr[dst];
    const int end = rowptr[dst + 1];
    float s = dis[dst];
    float acc = s * s * in[(long long)dst * D_IN + t];

    __shared__ int   ssrc[128];
    __shared__ float snorm[128];
    for (int base = beg; base < end; base += 128) {
        int m = end - base;
        if (m > 128) m = 128;
        __syncthreads();
        if (t < m) { ssrc[t] = esrc[base + t]; snorm[t] = enorm[base + t]; }
        __syncthreads();
        int j = 0;
        for (; j + 4 <= m; j += 4) {
            int s0 = ssrc[j], s1 = ssrc[j+1], s2 = ssrc[j+2], s3 = ssrc[j+3];
            float a0 = in[(long long)s0 * D_IN + t];
            float a1 = in[(long long)s1 * D_IN + t];
            float a2 = in[(long long)s2 * D_IN + t];
            float a3 = in[(long long)s3 * D_IN + t];
            acc += snorm[j]   * a0;
            acc += snorm[j+1] * a1;
            acc += snorm[j+2] * a2;
            acc += snorm[j+3] * a3;
        }
        for (; j < m; ++j)
            acc += snorm[j] * in[(long long)ssrc[j] * D_IN + t];
    }
    out[(long long)dst * D_IN + t] = acc;
}

// ------------------------------------------- SGConv linear via WMMA f32 16x16x4
__global__ void __launch_bounds__(128)
k_sgc_linear(const float* __restrict__ H,
             const float* __restrict__ Wc,
             const float* __restrict__ bc,
             float* __restrict__ hid) {
    __shared__ float sW[D_HID * D_IN];   // 32 KB: W_conv row-major (32 x 256)
    for (int i = threadIdx.x; i < D_HID * D_IN; i += 128) sW[i] = Wc[i];
    __syncthreads();

    const int wave = threadIdx.x >> 5;
    const int lane = threadIdx.x & 31;
    const int m0 = (blockIdx.x * 4 + wave) * 16;
    if (m0 >= N_NODES) return;           // wave-uniform; EXEC stays all-ones

    const int lh = lane & 15;
    const int hi = lane >> 4;

    const float* arow = H + (long long)(m0 + lh) * D_IN;
    const float* b0p = &sW[lh * D_IN];          // B cols 0..15 (B[k][n]=Wc[n][k])
    const float* b1p = &sW[(16 + lh) * D_IN];   // B cols 16..31

    v8f acc0 = {};
    v8f acc1 = {};
    for (int k = 0; k < D_IN; k += 4) {
        const int kk = k + 2 * hi;
        v2f a  = *(const v2f*)(arow + kk);
        v2f b0 = *(const v2f*)(b0p + kk);
        v2f b1 = *(const v2f*)(b1p + kk);
        acc0 = __builtin_amdgcn_wmma_f32_16x16x4_f32(false, a, false, b0,
                                                     (short)0, acc0, false, false);
        acc1 = __builtin_amdgcn_wmma_f32_16x16x4_f32(false, a, false, b1,
                                                     (short)0, acc1, false, false);
    }

    const float bias0 = bc[lh];
    const float bias1 = bc[16 + lh];
#pragma unroll
    for (int j = 0; j < 8; ++j) {
        int r = m0 + j + 8 * hi;
        float v0 = acc0[j] + bias0;
        float v1 = acc1[j] + bias1;
        v0 = (v0 > 0.0f) ? v0 : (expf(v0) - 1.0f);   // ELU(alpha=1)
        v1 = (v1 > 0.0f) ? v1 : (expf(v1) - 1.0f);
        hid[(long long)r * D_HID + lh]      = v0;
        hid[(long long)r * D_HID + 16 + lh] = v1;
    }
}

// ------------------------------------------- classifier + log_softmax
__global__ void __launch_bounds__(256)
k_fc_logsoftmax(const float* __restrict__ hid,
                const float* __restrict__ Wfc,
                const float* __restrict__ bfc,
                float* __restrict__ out) {
    __shared__ float sW[N_CLS * D_HID];
    __shared__ float sB[N_CLS];
    for (int i = threadIdx.x; i < N_CLS * D_HID; i += 256) sW[i] = Wfc[i];
    if (threadIdx.x < N_CLS) sB[threadIdx.x] = bfc[threadIdx.x];
    __syncthreads();

    int n = blockIdx.x * 256 + threadIdx.x;
    if (n >= N_NODES) return;

    float h[D_HID];
#pragma unroll
    for (int j = 0; j < D_HID; j += 4) {
        float4 v = *(const float4*)(hid + (long long)n * D_HID + j);
        h[j] = v.x; h[j + 1] = v.y; h[j + 2] = v.z; h[j + 3] = v.w;
    }

    float logits[N_CLS];
    float mx = -1e30f;
#pragma unroll 4
    for (int c = 0; c < N_CLS; ++c) {
        float acc = sB[c];
#pragma unroll
        for (int j = 0; j < D_HID; ++j) acc += h[j] * sW[c * D_HID + j];
        logits[c] = acc;
        mx = fmaxf(mx, acc);
    }
    float se = 0.0f;
#pragma unroll 4
    for (int c = 0; c < N_CLS; ++c) se += expf(logits[c] - mx);
    const float lse = mx + logf(se);
#pragma unroll 4
    for (int c = 0; c < N_CLS; ++c)
        out[(long long)n * N_CLS + c] = logits[c] - lse;
}

// ----------------------------------------------------------------------------
extern "C" void kernel_launch(void* const* d_in, const int* in_sizes, int n_in,
                              void* d_out, int out_size, void* d_ws, size_t ws_size,
                              hipStream_t stream) {
    (void)in_sizes; (void)n_in; (void)out_size; (void)ws_size;

    const float*     x   = (const float*)d_in[0];
    const long long* ei  = (const long long*)d_in[1];   // int64 (2, E)
    const float*     ea  = (const float*)d_in[2];       // (E, 1)
    const float*     Wc  = (const float*)d_in[3];       // (32, 256)
    const float*     bc  = (const float*)d_in[4];       // (32,)
    const float*     Wfc = (const float*)d_in[5];       // (40, 32)
    const float*     bfc = (const float*)d_in[6];       // (40,)
    float* out = (float*)d_out;                         // (N, 40)

    const long long* row = ei;            // edge_index[0]
    const long long* col = ei + N_EDGES;  // edge_index[1]

    // workspace layout, 4-byte units, each section 256B-aligned (P = padded N)
    constexpr size_t P = 100032;   // 100000 -> multiple of 64
    float* wsf = (float*)d_ws;
    int*   wsi = (int*)d_ws;
    float* deg    = wsf;                      // P floats
    float* dis    = wsf + P;                  // P floats
    int*   cnt    = wsi + 2 * P;              // P ints
    int*   rowptr = wsi + 3 * P;              // P ints (needs N+1)
    int*   pos    = wsi + 4 * P;              // P ints
    int*   bsum   = wsi + 5 * P;              // 512 ints
    int*   esrc   = wsi + 5 * P + 512;        // E ints
    float* enorm  = wsf + 5 * P + 512 + (size_t)N_EDGES;   // E floats
    float* Ha     = enorm + (size_t)N_EDGES;               // N*256
    float* Hb     = Ha + (size_t)N_NODES * D_IN;           // N*256
    float* hid    = enorm;   // reuse: enorm dead after hop 3; N*32 == E floats

    const int gN = (N_NODES + 255) / 256;   // 391
    const int gE = N_EDGES / 256;           // 12500

    // gcn_norm degrees + CSR-by-destination build
    k_init      <<<gN, 256, 0, stream>>>(deg, cnt);
    k_acc       <<<gE, 256, 0, stream>>>(col, ea, deg, cnt);
    k_dis       <<<gN, 256, 0, stream>>>(deg, dis);
    k_blocksum  <<<gN, 256, 0, stream>>>(cnt, bsum);
    k_scanblocks<<<1, 512, 0, stream>>>(bsum);
    k_rowptr    <<<gN, 256, 0, stream>>>(cnt, bsum, rowptr, pos);
    k_fill      <<<gE, 256, 0, stream>>>(row, col, ea, dis, pos, esrc, enorm);

    // 3 diffusion hops: pure gather, no atomics (H stays L2-resident: 102 MB)
    k_hop<<<N_NODES, 256, 0, stream>>>(rowptr, esrc, enorm, dis, x,  Ha);
    k_hop<<<N_NODES, 256, 0, stream>>>(rowptr, esrc, enorm, dis, Ha, Hb);
    k_hop<<<N_NODES, 256, 0, stream>>>(rowptr, esrc, enorm, dis, Hb, Ha);

    // SGConv linear (WMMA f32) + ELU -> hid
    k_sgc_linear<<<(N_NODES / 16 + 3) / 4, 128, 0, stream>>>(Ha, Wc, bc, hid);

    // classifier + log_softmax -> out
    k_fc_logsoftmax<<<gN, 256, 0, stream>>>(hid, Wfc, bfc, out);
}